// GauntProductBasis_74723841016344
// MI455X (gfx1250) — compile-verified
//
#include <hip/hip_runtime.h>
#include <hip/hip_bf16.h>
#include <math.h>

// ---------------------------------------------------------------------------
// Gaunt product basis, fully fused as WMMA f32 GEMMs on gfx1250 (wave32).
//   stage 1: x_f = A x            (16 -> 722 reals, via V_WMMA_F32_16X16X4_F32)
//   stage 2: p2 = x_f^2, p3 = x_f^3   (elementwise in D-layout registers)
//   stage 3: y  = L1 x + C [p2re;p2im;p3re;p3im]   (1444 -> 16, WMMA)
// Constant matrices are built on-device from M_re/M_im/S_re/S_im/path_w.
// ---------------------------------------------------------------------------

typedef float v2f __attribute__((ext_vector_type(2)));
typedef float v8f __attribute__((ext_vector_type(8)));

#define NROWS   131072      // N * MUL
#define DIN     16
#define DOUT    16
#define GRID    19          // SIZE
#define WREAL   361         // 19*19
#define WPAD    368         // padded to 23 tiles of 16
#define NWT     23
#define TWO_PI_19 0.3306939635357677f  /* 2*pi/19 */

// ws layout (floats):
#define B_OFF   0                       // Bpack: 2 (re/im) * 8 pairs * 368 w * 2  = 11776
#define C_OFF   11776                   // Cpack: 4 q * 184 pairs * 16 j * 2       = 23552
#define L_OFF   (11776 + 23552)         // Lpack: 8 pairs * 16 j * 2               = 256
#define WS_FLOATS (L_OFF + 256)

static __device__ inline v8f wmma4(v2f a, v2f b, v8f c) {
  return __builtin_amdgcn_wmma_f32_16x16x4_f32(
      /*neg_a=*/false, a, /*neg_b=*/false, b,
      /*c_mod=*/(short)0, c, /*reuse_a=*/false, /*reuse_b=*/false);
}

// --------------------------- precompute kernels ----------------------------

// Bpack[c][p][w] = (B[c][2p][w], B[c][2p+1][w]),  B = A^T of the fused
// (M -> zero-padded ortho FFT) map:  A[w,i] = (1/19) sum_{u,v<7} M[i,u,v] e^{-i th}
__global__ void pre_B(const float* __restrict__ Mre, const float* __restrict__ Mim,
                      float* __restrict__ Bpack) {
  int w = blockIdx.x;            // 0..367
  int t = threadIdx.x;           // 0..31
  int i = t & 15, c = t >> 4;    // c: 0 = re, 1 = im
  float s = 0.f;
  if (w < WREAL) {
    int wu = w / GRID, wv = w % GRID;
    for (int u = 0; u < 7; ++u)
      for (int v = 0; v < 7; ++v) {
        int ph = (wu * u + wv * v) % GRID;
        float th = (float)ph * TWO_PI_19;
        float ct = cosf(th), st = sinf(th);
        float a = Mre[(i * 7 + u) * 7 + v];
        float b = Mim[(i * 7 + u) * 7 + v];
        // (a+ib) * (cos - i sin):  re = a c + b s ; im = b c - a s
        s += (c == 0) ? (a * ct + b * st) : (b * ct - a * st);
      }
    s *= (1.f / 19.f);
  }
  Bpack[((c * 8 + (i >> 1)) * WPAD + w) * 2 + (i & 1)] = s;
}

// Cpack[q][p][j]: q0=G2re, q1=-G2im, q2=G3re, q3=-G3im (all * path_w[j]/19),
// packed in (w-pair, j) order: element (q, w/2, j) = (C[q][w][j], C[q][w+1][j]).
__global__ void pre_C(const float* __restrict__ Sre, const float* __restrict__ Sim,
                      const float* __restrict__ pw, float* __restrict__ Cpack) {
  int w = blockIdx.x;      // 0..367
  int j = threadIdx.x;     // 0..15
  float g2r = 0.f, g2i = 0.f, g3r = 0.f, g3i = 0.f;
  if (w < WREAL) {
    int wu = w / GRID, wv = w % GRID;
    // v=2 window: conv[t] placed at S[j][t+3], t in [0,13)
    for (int tu = 0; tu < 13; ++tu)
      for (int tv = 0; tv < 13; ++tv) {
        int ph = (wu * tu + wv * tv) % GRID;
        float th = (float)ph * TWO_PI_19;
        float ct = cosf(th), st = sinf(th);
        float a = Sre[(j * GRID + tu + 3) * GRID + tv + 3];
        float b = Sim[(j * GRID + tu + 3) * GRID + tv + 3];
        // (a+ib) * (cos + i sin):  re = a c - b s ; im = b c + a s
        g2r += a * ct - b * st;
        g2i += b * ct + a * st;
      }
    // v=3 window: full 19x19
    for (int tu = 0; tu < GRID; ++tu)
      for (int tv = 0; tv < GRID; ++tv) {
        int ph = (wu * tu + wv * tv) % GRID;
        float th = (float)ph * TWO_PI_19;
        float ct = cosf(th), st = sinf(th);
        float a = Sre[(j * GRID + tu) * GRID + tv];
        float b = Sim[(j * GRID + tu) * GRID + tv];
        g3r += a * ct - b * st;
        g3i += b * ct + a * st;
      }
  }
  float scale = pw[j] * (1.f / 19.f);
  float q[4] = { g2r * scale, -g2i * scale, g3r * scale, -g3i * scale };
  for (int qi = 0; qi < 4; ++qi)
    Cpack[((qi * (WPAD / 2) + (w >> 1)) * 16 + j) * 2 + (w & 1)] = q[qi];
}

// Lpack: linear term, Re( sum_{u,v<7} M[i,u,v] * S[j,u+6,v+6] ) * pw[j],
// packed as (i-pair, j).
__global__ void pre_L(const float* __restrict__ Mre, const float* __restrict__ Mim,
                      const float* __restrict__ Sre, const float* __restrict__ Sim,
                      const float* __restrict__ pw, float* __restrict__ Lpack) {
  int i = threadIdx.x;   // 16
  int j = threadIdx.y;   // 16
  float s = 0.f;
  for (int u = 0; u < 7; ++u)
    for (int v = 0; v < 7; ++v) {
      s += Mre[(i * 7 + u) * 7 + v] * Sre[(j * GRID + u + 6) * GRID + v + 6]
         - Mim[(i * 7 + u) * 7 + v] * Sim[(j * GRID + u + 6) * GRID + v + 6];
    }
  s *= pw[j];
  Lpack[((i >> 1) * 16 + j) * 2 + (i & 1)] = s;
}

// ------------------------------ main kernel --------------------------------

__global__ __launch_bounds__(256, 1)
void gaunt_main(const float* __restrict__ X,
                const float* __restrict__ Bp,
                const float* __restrict__ Cp,
                const float* __restrict__ Lp,
                float* __restrict__ Y) {
  // per-wave LDS staging: 4 quantities * 16 rows * stride 18 floats
  __shared__ float lds[8 * 4 * 16 * 18];   // 36,864 B

  const int tid  = threadIdx.x;
  const int wave = tid >> 5;
  const int lane = tid & 31;
  const int half = lane >> 4;      // K-half selector for A/B layouts
  const int lr   = lane & 15;      // row (A) / column (B,D) index

  const long row0 = ((long)blockIdx.x * 8 + wave) * 16;

  const v2f* B2 = (const v2f*)Bp;  // [2][8][WPAD]  pairs over i
  const v2f* C2 = (const v2f*)Cp;  // [4][184][16]  pairs over w
  const v2f* L2 = (const v2f*)Lp;  // [8][16]       pairs over i

  // ---- load X tile as four A-chunks (K = 4*kk + 2*half + {0,1}) ----
  v2f xa[4];
  const float* xrow = X + (row0 + lr) * DIN;
#pragma unroll
  for (int kk = 0; kk < 4; ++kk)
    xa[kk] = *(const v2f*)(xrow + 4 * kk + 2 * half);

  v8f acc = {0.f, 0.f, 0.f, 0.f, 0.f, 0.f, 0.f, 0.f};

  // ---- linear term: acc += X * L1 ----
#pragma unroll
  for (int kk = 0; kk < 4; ++kk)
    acc = wmma4(xa[kk], L2[(2 * kk + half) * 16 + lr], acc);

  float* myl = lds + wave * (4 * 16 * 18);

  for (int wt = 0; wt < NWT; ++wt) {
    const int w0 = wt * 16;

    // ---- GEMM1: x_f tile (re, im) in D layout ----
    v8f fre = {0.f, 0.f, 0.f, 0.f, 0.f, 0.f, 0.f, 0.f};
    v8f fim = {0.f, 0.f, 0.f, 0.f, 0.f, 0.f, 0.f, 0.f};
#pragma unroll
    for (int kk = 0; kk < 4; ++kk) {
      fre = wmma4(xa[kk], B2[(    2 * kk + half) * WPAD + w0 + lr], fre);
      fim = wmma4(xa[kk], B2[(8 + 2 * kk + half) * WPAD + w0 + lr], fim);
    }

    // ---- elementwise complex square/cube, stage to LDS (D -> A transpose) ----
#pragma unroll
    for (int e = 0; e < 8; ++e) {
      float fr = fre[e], fi = fim[e];
      float p2r = fr * fr - fi * fi;
      float p2i = 2.f * fr * fi;
      float p3r = p2r * fr - p2i * fi;
      float p3i = p2r * fi + p2i * fr;
      int off = (e + 8 * half) * 18 + lr;   // row m = e + 8*half, col n = lr
      myl[0 * 288 + off] = p2r;
      myl[1 * 288 + off] = p2i;
      myl[2 * 288 + off] = p3r;
      myl[3 * 288 + off] = p3i;
    }
    // LDS ops from the same wave are in-order; register deps insert s_wait_dscnt.

    // ---- GEMM2: acc += P_q * C_q over this w-tile ----
#pragma unroll
    for (int q = 0; q < 4; ++q) {
#pragma unroll
      for (int kk = 0; kk < 4; ++kk) {
        v2f a = *(const v2f*)(myl + q * 288 + lr * 18 + 4 * kk + 2 * half);
        v2f b = C2[(q * (WPAD / 2) + wt * 8 + 2 * kk + half) * 16 + lr];
        acc = wmma4(a, b, acc);
      }
    }
  }

  // ---- store D-layout accumulator ----
  float* yrow = Y + row0 * DOUT;
#pragma unroll
  for (int e = 0; e < 8; ++e)
    yrow[(e + 8 * half) * DOUT + lr] = acc[e];
}

// ------------------------------ launcher -----------------------------------

extern "C" void kernel_launch(void* const* d_in, const int* in_sizes, int n_in,
                              void* d_out, int out_size, void* d_ws, size_t ws_size,
                              hipStream_t stream) {
  const float* x   = (const float*)d_in[0];
  const float* Mre = (const float*)d_in[1];
  const float* Mim = (const float*)d_in[2];
  const float* Sre = (const float*)d_in[3];
  const float* Sim = (const float*)d_in[4];
  const float* pw  = (const float*)d_in[5];

  float* ws = (float*)d_ws;
  float* Bp = ws + B_OFF;
  float* Cp = ws + C_OFF;
  float* Lp = ws + L_OFF;

  pre_B<<<WPAD, 32, 0, stream>>>(Mre, Mim, Bp);
  pre_C<<<WPAD, 16, 0, stream>>>(Sre, Sim, pw, Cp);
  dim3 lb(16, 16);
  pre_L<<<1, lb, 0, stream>>>(Mre, Mim, Sre, Sim, pw, Lp);

  gaunt_main<<<NROWS / (8 * 16), 256, 0, stream>>>(x, Bp, Cp, Lp, (float*)d_out);
}